// TenNet_34866544509652
// MI455X (gfx1250) — compile-verified
//
#include <hip/hip_runtime.h>
#include <hip/hip_bf16.h>

#define D_  10
#define N_  500
#define MH  5      // hidden width
#define LB  1024   // batch

typedef __attribute__((ext_vector_type(2))) float v2f;
typedef __attribute__((ext_vector_type(8))) float v8f;

#define LOG2E_F   1.4426950408889634f
#define LN2_F     0.6931471805599453f

// branchless native helpers: single v_exp_f32 / v_log_f32 / v_rcp_f32, no EXEC juggling
__device__ __forceinline__ float nexp2(float x)  { return __builtin_amdgcn_exp2f(x); }
__device__ __forceinline__ float nlog2(float x)  { return __builtin_amdgcn_logf(x); }
__device__ __forceinline__ float nrcp(float x)   { return __builtin_amdgcn_rcpf(x); }

// tanh(x) = sign(x) * (1 - 2/(exp(2|x|)+1)); exp2->inf gives exactly 1
__device__ __forceinline__ float fast_tanh(float x) {
    float ax = fabsf(x);
    float e  = nexp2((2.0f * LOG2E_F) * ax);
    float t  = 1.0f - 2.0f * nrcp(e + 1.0f);
    return copysignf(t, x);
}

// sigmoid(x) = 1/(1+exp(-x))
__device__ __forceinline__ float fast_sigmoid(float x) {
    return nrcp(1.0f + nexp2(-LOG2E_F * x));
}

// softplus10(x) = max(x,0) + 0.1*log1p(exp(-10|x|)); e<=1 so 1+e form is safe
__device__ __forceinline__ float sp10(float x) {
    float e = nexp2((-10.0f * LOG2E_F) * fabsf(x));
    return fmaxf(x, 0.f) + (0.1f * LN2_F) * nlog2(1.0f + e);
}

// broadcast value from lane `srcLane` (0..31) to this lane (wave32)
__device__ __forceinline__ float bcast_lane(float v, int srcLane) {
    return __int_as_float(__builtin_amdgcn_ds_bpermute(srcLane << 2, __float_as_int(v)));
}

// One block per (d,n). 8 waves; each wave runs 8 tiles of 16 batch rows.
// WMMA orientation: C = W^T (A, 16x4) x phis^T (B, 4x16):
//   A lanes 0-15 hold K=0,1 (VGPR .x,.y), lanes 16-31 hold K=2,3; M = out-feature = lane&15
//   B lanes 0-15 hold K=0,1 rows (N = batch col = lane), lanes 16-31 hold K=2,3
//   C VGPR r: lanes 0-15 -> (M=r, N=lane); features land in fixed VGPR index.
__global__ __launch_bounds__(256) void tennet_main(
    const float* __restrict__ X,       // [1024,10]
    const float* __restrict__ w_first, // [10,500,1,5]
    const float* __restrict__ w_mid,   // [4,10,500,5,5]
    const float* __restrict__ w_last,  // [10,500,5,1]
    const float* __restrict__ bs,      // [5,10,500,5]
    const float* __restrict__ b_last,  // [10,500,1]
    const float* __restrict__ a_all,   // [5,10,500,5]
    float* __restrict__ out)           // [1024,10,500]
{
    const int dn   = blockIdx.x;           // 0..4999
    const int d    = dn / N_;
    const int n    = dn % N_;
    const int lane = threadIdx.x & 31;
    const int wave = threadIdx.x >> 5;
    const int f    = lane & 15;            // out-feature (A) / batch col (B,C)
    const bool lo  = lane < 16;

    // ---------- per-(d,n) parameter prep (amortized over 1024 batch rows) ----------
    float W0[MH], b0[MH], ta0[MH];
    {
        const float* wf = w_first + (size_t)dn * MH;
        const float* bb = bs      + (size_t)dn * MH;   // layer l=0 slice
        const float* aa = a_all   + (size_t)dn * MH;
#pragma unroll
        for (int k = 0; k < MH; ++k) {
            W0[k]  = sp10(wf[k]);
            b0[k]  = bb[k];
            ta0[k] = fast_tanh(aa[k]);
        }
    }

    // Mid layers: prebuild WMMA A operands (A = softplus10(W)^T, zero-padded to 16x8)
    float a1x[4], a1y[4], a2x[4];
    float bias[4][MH], tam[4][MH];
#pragma unroll
    for (int lm = 0; lm < 4; ++lm) {
        const float* Wl = w_mid + ((size_t)(lm * D_ + d) * N_ + n) * (MH * MH); // [k][f]
        const int k0 = lo ? 0 : 2;
        const int k1 = lo ? 1 : 3;
        float v0 = 0.f, v1 = 0.f, v2 = 0.f;
        if (f < MH) {                        // guard: stay inside this (d,n)'s 25-block
            v0 = sp10(Wl[k0 * MH + f]);
            v1 = sp10(Wl[k1 * MH + f]);
            if (lo) v2 = sp10(Wl[4 * MH + f]);   // K=4 (second WMMA); K=5..7 are zero
        }
        a1x[lm] = v0; a1y[lm] = v1; a2x[lm] = v2;

        const size_t idx = ((size_t)((lm + 1) * D_ + d) * N_ + n) * MH;
#pragma unroll
        for (int k = 0; k < MH; ++k) {
            bias[lm][k] = bs[idx + k];
            tam [lm][k] = fast_tanh(a_all[idx + k]);
        }
    }

    float WL[MH], bL;
    {
        const float* wl = w_last + (size_t)dn * MH;
#pragma unroll
        for (int k = 0; k < MH; ++k) WL[k] = sp10(wl[k]);
        bL = b_last[dn];
    }

    // ---------- batch tiles ----------
    for (int t = 0; t < 8; ++t) {
        const int m = ((wave * 8 + t) << 4) + f;   // lanes 16-31 mirror lanes 0-15
        const float x = X[(size_t)m * D_ + d];

        // layer 0: scalar -> 5  (phidots starts at 1, so matvec is just W0[f])
        float phi[MH], phd[MH];
#pragma unroll
        for (int k = 0; k < MH; ++k) {
            float pre  = fmaf(x, W0[k], b0[k]);
            float th   = fast_tanh(pre);
            float tdot = 1.f - th * th;
            phd[k] = W0[k] * fmaf(ta0[k], tdot, 1.f);
            phi[k] = fmaf(th, ta0[k], pre);
        }

        // mid layers: 5->5 via WMMA (2 WMMAs for phis, 2 for phidots)
#pragma unroll
        for (int lm = 0; lm < 4; ++lm) {
            v2f A1; A1.x = a1x[lm]; A1.y = a1y[lm];
            v2f A2; A2.x = a2x[lm]; A2.y = 0.f;

            // build B = vec^T in 4x16 layout: upper lanes need rows K=2,3 from lanes 0-15
            float p2 = bcast_lane(phi[2], f), p3 = bcast_lane(phi[3], f);
            float q2 = bcast_lane(phd[2], f), q3 = bcast_lane(phd[3], f);
            v2f Bp1; Bp1.x = lo ? phi[0] : p2;  Bp1.y = lo ? phi[1] : p3;
            v2f Bp2; Bp2.x = lo ? phi[4] : 0.f; Bp2.y = 0.f;
            v2f Bq1; Bq1.x = lo ? phd[0] : q2;  Bq1.y = lo ? phd[1] : q3;
            v2f Bq2; Bq2.x = lo ? phd[4] : 0.f; Bq2.y = 0.f;

            v8f Cp = {};
#pragma unroll
            for (int k = 0; k < MH; ++k) Cp[k] = bias[lm][k];   // bias folded into accum
            v8f Cq = {};

            Cp = __builtin_amdgcn_wmma_f32_16x16x4_f32(false, A1, false, Bp1, (short)0, Cp, false, false);
            Cp = __builtin_amdgcn_wmma_f32_16x16x4_f32(false, A2, false, Bp2, (short)0, Cp, false, false);
            Cq = __builtin_amdgcn_wmma_f32_16x16x4_f32(false, A1, false, Bq1, (short)0, Cq, false, false);
            Cq = __builtin_amdgcn_wmma_f32_16x16x4_f32(false, A2, false, Bq2, (short)0, Cq, false, false);

#pragma unroll
            for (int k = 0; k < MH; ++k) {
                float pre  = Cp[k];
                float th   = fast_tanh(pre);
                float tdot = 1.f - th * th;
                phd[k] = Cq[k] * fmaf(tam[lm][k], tdot, 1.f);
                phi[k] = fmaf(th, tam[lm][k], pre);
            }
        }

        // last layer: 5 -> 1, sigmoid gate
        float pre = bL, pd = 0.f;
#pragma unroll
        for (int k = 0; k < MH; ++k) {
            pre = fmaf(phi[k], WL[k], pre);
            pd  = fmaf(phd[k], WL[k], pd);
        }
        float s = fast_sigmoid(pre);
        pd = pd * s * (1.f - s);

        if (lo) out[(size_t)m * (D_ * N_) + dn] = pd;   // after WMMAs; EXEC restored next iter
    }
}

// fm[m] = min_n ( (1/D) * sum_d -log(phidot[m,d,n] + 1e-10) )
__global__ __launch_bounds__(256) void tennet_fm(const float* __restrict__ pd,
                                                 float* __restrict__ fm)
{
    const int m   = blockIdx.x;
    const int tid = threadIdx.x;
    __shared__ float red[256];

    float best = 3.4e38f;
    for (int n = tid; n < N_; n += 256) {
        const float* row = pd + (size_t)m * (D_ * N_) + n;
        float s = 0.f;
#pragma unroll
        for (int dd = 0; dd < D_; ++dd) s -= nlog2(row[dd * N_] + 1e-10f);
        best = fminf(best, s * (LN2_F / D_));   // fold ln2 and 1/D into one scale
    }
    red[tid] = best;
    __syncthreads();
#pragma unroll
    for (int off = 128; off > 0; off >>= 1) {
        if (tid < off) red[tid] = fminf(red[tid], red[tid + off]);
        __syncthreads();
    }
    if (tid == 0) fm[m] = red[0];
}

extern "C" void kernel_launch(void* const* d_in, const int* in_sizes, int n_in,
                              void* d_out, int out_size, void* d_ws, size_t ws_size,
                              hipStream_t stream)
{
    const float* X       = (const float*)d_in[0];
    const float* w_first = (const float*)d_in[1];
    const float* w_mid   = (const float*)d_in[2];
    const float* w_last  = (const float*)d_in[3];
    const float* bs      = (const float*)d_in[4];
    const float* b_last  = (const float*)d_in[5];
    const float* a_all   = (const float*)d_in[6];

    float* out   = (float*)d_out;
    float* pdout = out;                               // [1024,10,500]
    float* fmout = out + (size_t)LB * D_ * N_;        // [1024]

    tennet_main<<<dim3(D_ * N_), dim3(256), 0, stream>>>(
        X, w_first, w_mid, w_last, bs, b_last, a_all, pdout);
    tennet_fm<<<dim3(LB), dim3(256), 0, stream>>>(pdout, fmout);
}